// SeqCEL_18021682774493
// MI455X (gfx1250) — compile-verified
//
#include <hip/hip_runtime.h>
#include <hip/hip_bf16.h>
#include <cstdint>

// Problem constants (from reference): preds[B,T,V] f32, targets[B,T] int,
// label_sizes[B] int, scalar f32 output = sum of masked NLL.
#define B_DIM 16
#define T_DIM 512
#define V_DIM 32000
#define ROWS  (B_DIM * T_DIM)          // 8192

// Streaming parameters: one wave32 per row.
#define WAVES_PER_BLOCK 8
#define CHUNK_FLOATS 128               // 32 lanes x float4
#define CHUNK_BYTES  512
#define NCHUNK (V_DIM / CHUNK_FLOATS)  // 250 chunks per row (exact)
#define DEPTH 4                        // async pipeline depth (ring buffers)

// Phase 1: one wave streams one row (128 KB) through a private LDS ring using
// CDNA5 async global->LDS copies (ASYNCcnt, non-temporal), computing an
// online (single-pass) log-sum-exp in base 2.
__global__ __launch_bounds__(256) void SeqCEL_rows_kernel(
    const float* __restrict__ preds,
    const int*   __restrict__ targets,
    const int*   __restrict__ label_sizes,
    float*       __restrict__ row_loss)
{
    // 8 waves * 4 ring slots * 512B = 16 KB LDS per workgroup.
    __shared__ float lds[WAVES_PER_BLOCK * DEPTH * CHUNK_FLOATS];

    const int lane = threadIdx.x & 31;
    const int wave = threadIdx.x >> 5;
    const int row  = blockIdx.x * WAVES_PER_BLOCK + wave;   // grid sized exactly

    const float* rowPtr = preds + (long long)row * V_DIM;

    float* mybuf = &lds[wave * DEPTH * CHUNK_FLOATS];
    // Generic pointer to __shared__ is {aperture_hi, lds_offset}; low 32 bits
    // are the LDS byte address the async-DMA VDST field expects.
    const uint32_t lds0 = (uint32_t)(uintptr_t)(void*)mybuf + (uint32_t)lane * 16u;
    const uint64_t g0   = (uint64_t)(uintptr_t)rowPtr + (uint64_t)((uint32_t)lane * 16u);

    // Prologue: prime the pipeline with DEPTH-1 chunks in flight.
    for (int c = 0; c < DEPTH - 1; ++c) {
        uint32_t ldsAddr = lds0 + (uint32_t)c * CHUNK_BYTES;
        uint64_t gAddr   = g0   + (uint64_t)c * CHUNK_BYTES;
        asm volatile("global_load_async_to_lds_b128 %0, %1, off th:TH_LOAD_NT"
                     :: "v"(ldsAddr), "v"(gAddr) : "memory");
    }

    const float LOG2E = 1.4426950408889634f;
    float m = -3.0e38f;   // running max of x*log2(e)
    float s = 0.0f;       // running sum of 2^(x*log2e - m)

    for (int i = 0; i < NCHUNK; ++i) {
        const int nx = i + (DEPTH - 1);
        if (nx < NCHUNK) {
            uint32_t ldsAddr = lds0 + (uint32_t)(nx & (DEPTH - 1)) * CHUNK_BYTES;
            uint64_t gAddr   = g0   + (uint64_t)nx * CHUNK_BYTES;
            asm volatile("global_load_async_to_lds_b128 %0, %1, off th:TH_LOAD_NT"
                         :: "v"(ldsAddr), "v"(gAddr) : "memory");
            // Loads retire in order: <=3 outstanding means chunk i is in LDS.
            asm volatile("s_wait_asynccnt 3" ::: "memory");
        } else {
            asm volatile("s_wait_asynccnt 0" ::: "memory");
        }

        const float4 v4 = *reinterpret_cast<const float4*>(
            &mybuf[(i & (DEPTH - 1)) * CHUNK_FLOATS + lane * 4]);

        // Branchless online-softmax update per float4 (no EXEC divergence):
        // rescale the running sum once for the 4-wide max, then add 4 exps.
        const float t0 = v4.x * LOG2E;
        const float t1 = v4.y * LOG2E;
        const float t2 = v4.z * LOG2E;
        const float t3 = v4.w * LOG2E;
        const float m4 = fmaxf(fmaxf(t0, t1), fmaxf(t2, t3));
        const float Mn = fmaxf(m, m4);
        s = s * exp2f(m - Mn)
          + ((exp2f(t0 - Mn) + exp2f(t1 - Mn)) +
             (exp2f(t2 - Mn) + exp2f(t3 - Mn)));
        m = Mn;
    }

    // Wave32 combine of (max, scaled-sum) pairs.
    #pragma unroll
    for (int off = 16; off > 0; off >>= 1) {
        const float m2 = __shfl_xor(m, off, 32);
        const float s2 = __shfl_xor(s, off, 32);
        const float M  = fmaxf(m, m2);
        s = s * exp2f(m - M) + s2 * exp2f(m2 - M);
        m = M;
    }

    if (lane == 0) {
        const int b = row >> 9;        // T = 512
        const int t = row & (T_DIM - 1);
        const int tgt = targets[row];
        const float xt  = rowPtr[tgt];
        const float lse = (m + log2f(s)) * 0.6931471805599453f;   // back to ln
        const float nll = lse - xt;
        row_loss[row] = (t < label_sizes[b]) ? nll : 0.0f;
    }
}

// Phase 2: deterministic fixed-order reduction of the 8192 per-row losses.
__global__ __launch_bounds__(256) void SeqCEL_reduce_kernel(
    const float* __restrict__ row_loss, float* __restrict__ out)
{
    __shared__ float sm[256];
    float acc = 0.0f;
    for (int i = threadIdx.x; i < ROWS; i += 256) acc += row_loss[i];
    sm[threadIdx.x] = acc;
    __syncthreads();
    for (int st = 128; st > 0; st >>= 1) {
        if ((int)threadIdx.x < st) sm[threadIdx.x] += sm[threadIdx.x + st];
        __syncthreads();
    }
    if (threadIdx.x == 0) out[0] = sm[0];
}

extern "C" void kernel_launch(void* const* d_in, const int* in_sizes, int n_in,
                              void* d_out, int out_size, void* d_ws, size_t ws_size,
                              hipStream_t stream) {
    const float* preds       = (const float*)d_in[0];
    const int*   targets     = (const int*)d_in[1];
    const int*   label_sizes = (const int*)d_in[2];
    float*       out         = (float*)d_out;
    float*       row_loss    = (float*)d_ws;   // 8192 floats = 32 KB scratch

    SeqCEL_rows_kernel<<<ROWS / WAVES_PER_BLOCK, 256, 0, stream>>>(
        preds, targets, label_sizes, row_loss);
    SeqCEL_reduce_kernel<<<1, 256, 0, stream>>>(row_loss, out);
}